// CapsNet_69114613730132
// MI455X (gfx1250) — compile-verified
//
#include <hip/hip_runtime.h>
#include <math.h>

typedef __attribute__((ext_vector_type(2))) float v2f;
typedef __attribute__((ext_vector_type(8))) float v8f;

#define EPS 1e-7f

// Workspace layout (float offsets)
#define WS_H   0         // 128*256          = 32768   conv1 output
#define WS_U   32768     // 512*8            = 4096    squashed capsules
#define WS_A1  36864     // 128*256          = 32768   conv_w, K padded 243->256
#define WS_B1  69632     // 256*256          = 65536   im2col(x), pair-interleaved
#define WS_B2  135168    // 10368*16         = 165888  im2col(h), pair-interleaved
// total 301056 floats = 1.204 MB

// ---------------------------------------------------------------------------
// Staging for conv1: zero-padded A panel + pair-interleaved im2col(x) B panel.
//   A1[row*256 + k]                      (k >= 243 -> 0)
//   B1[(k>>1)*512 + n*2 + (k&1)] = im2col(x)[k][n]   (k >= 243 -> 0)
// ---------------------------------------------------------------------------
__global__ __launch_bounds__(256) void k_stage1(
    const float* __restrict__ x, const float* __restrict__ w,
    float* __restrict__ ws)
{
    const int gid = blockIdx.x * 256 + threadIdx.x;   // 98304 threads
    if (gid < 32768) {                                // A panel: 128 x 256
        const int row = gid >> 8, k = gid & 255;
        ws[WS_A1 + row * 256 + k] = (k < 243) ? w[row * 243 + k] : 0.0f;
    } else {                                          // B panel: 256 x 256
        const int idx = gid - 32768;
        const int k = idx >> 8, n = idx & 255;
        float v = 0.0f;
        if (k < 243) {
            const int ci = k / 81, rem = k - ci * 81, ky = rem / 9, kx = rem - ky * 9;
            const int y = n >> 4, xc = n & 15;
            v = x[ci * 576 + (y + ky) * 24 + (xc + kx)];
        }
        ws[WS_B1 + (k >> 1) * 512 + n * 2 + (k & 1)] = v;
    }
}

// ---------------------------------------------------------------------------
// Kernel 1: Conv1 GEMM.  M=128, K=256(padded), N=256.
//   128 single-wave blocks = 8 M-tiles x 16 N-tiles; N-tile == output row y.
//   Inner loop: 2x global_load_b64 + v_wmma_f32_16x16x4_f32, no branches.
// ---------------------------------------------------------------------------
__global__ __launch_bounds__(32) void k1_conv1(
    const float* __restrict__ ws, const float* __restrict__ bias,
    float* __restrict__ h)
{
    const int tile_m = blockIdx.x >> 4;
    const int y      = blockIdx.x & 15;
    const int lane   = threadIdx.x;
    const int half   = lane >> 4;
    const int r      = lane & 15;

    const float* arow = ws + WS_A1 + (tile_m * 16 + r) * 256 + 2 * half;
    const float* bcol = ws + WS_B1 + (y * 16 + r) * 2 + half * 512;

    v8f acc = {};
#pragma unroll 4
    for (int kk = 0; kk < 256; kk += 4) {
        const v2f a  = *(const v2f*)(arow + kk);
        const v2f bb = *(const v2f*)(bcol + (kk >> 1) * 512);
        acc = __builtin_amdgcn_wmma_f32_16x16x4_f32(false, a, false, bb,
                                                    (short)0, acc, false, false);
    }
#pragma unroll
    for (int v = 0; v < 8; ++v) {
        const int c = tile_m * 16 + v + 8 * half;          // D: M = v + 8*half
        h[c * 256 + y * 16 + r] = acc[v] + bias[c];        //    N = lane & 15
    }
}

// ---------------------------------------------------------------------------
// Staging for conv2: pair-interleaved im2col(h).
//   B2[(k>>1)*32 + n*2 + (k&1)] = h[ci][2py+ky][2px+kx],  k=ci*81+ky*9+kx
// ---------------------------------------------------------------------------
__global__ __launch_bounds__(256) void k_im2col2(
    const float* __restrict__ h, float* __restrict__ ws)
{
    const int gid = blockIdx.x * 256 + threadIdx.x;   // 165888 threads
    const int k = gid >> 4, n = gid & 15;
    const int ci = k / 81, rem = k - ci * 81, ky = rem / 9, kx = rem - ky * 9;
    const int py = n >> 2, px = n & 3;
    ws[WS_B2 + (k >> 1) * 32 + n * 2 + (k & 1)] =
        h[ci * 256 + (2 * py + ky) * 16 + (2 * px + kx)];
}

// ---------------------------------------------------------------------------
// Kernel 2: PrimaryCaps GEMM.  M=256, K=10368, N=16, split-K over 8 waves,
//           LDS reduction, fused bias + reshape(512,8) + squash.
// ---------------------------------------------------------------------------
__global__ __launch_bounds__(256) void k2_conv2_squash(
    const float* __restrict__ ws, const float* __restrict__ w,
    const float* __restrict__ bias, float* __restrict__ u)
{
    __shared__ float red[8 * 256 + 256];
    const int tid    = threadIdx.x;
    const int wave   = tid >> 5;
    const int lane   = tid & 31;
    const int half   = lane >> 4;
    const int r      = lane & 15;
    const int cobase = blockIdx.x * 16;

    const int   kbase = wave * 1296;                 // 8*1296 = 10368 = K
    const float* arow = w + (long)(cobase + r) * 10368 + kbase + 2 * half;
    const float* bcol = ws + WS_B2 + (kbase >> 1) * 32 + r * 2 + half * 32;

    v8f acc = {};
#pragma unroll 4
    for (int it = 0; it < 324; ++it) {
        const v2f a  = *(const v2f*)(arow + it * 4);
        const v2f bb = *(const v2f*)(bcol + it * 64);
        acc = __builtin_amdgcn_wmma_f32_16x16x4_f32(false, a, false, bb,
                                                    (short)0, acc, false, false);
    }
#pragma unroll
    for (int v = 0; v < 8; ++v)
        red[wave * 256 + (v + 8 * half) * 16 + r] = acc[v];
    __syncthreads();

    // Split-K reduction + bias: p[m][n], m = tid>>4 (channel), n = tid&15
    {
        float s = 0.0f;
#pragma unroll
        for (int wv = 0; wv < 8; ++wv) s += red[wv * 256 + tid];
        red[2048 + tid] = s + bias[cobase + (tid >> 4)];
    }
    __syncthreads();

    // Squash: capsule u[n] = 8 consecutive spatial slots of one channel.
    {
        const int nl = tid >> 3, j = tid & 7;
        const int sbase = 2048 + (nl >> 1) * 16 + ((nl & 1) << 3);
        float sq = 0.0f;
#pragma unroll
        for (int jj = 0; jj < 8; ++jj) { float t = red[sbase + jj]; sq += t * t; }
        const float pv = red[sbase + j];
        u[(blockIdx.x * 32 + nl) * 8 + j] =
            sq / (1.0f + sq) * (pv / (sqrtf(sq + EPS) + EPS));
    }
}

// ---------------------------------------------------------------------------
// Kernel 3: routing (cij == 1/512; bij never updated) + elementwise squash
//           (trailing squash axis has size 1).  One wave per output element.
// ---------------------------------------------------------------------------
__global__ __launch_bounds__(32) void k3_route(
    const float* __restrict__ W, const float* __restrict__ u,
    float* __restrict__ out)
{
    const int k = blockIdx.x >> 4;   // 0..9
    const int i = blockIdx.x & 15;   // 0..15
    const int lane = threadIdx.x;

    float part = 0.0f;
    for (int n = lane; n < 512; n += 32) {
        const float* wp = W + (((long)n * 10 + k) * 16 + i) * 8;
        const float* up = u + n * 8;
#pragma unroll
        for (int j = 0; j < 8; ++j) part += wp[j] * up[j];
    }
#pragma unroll
    for (int off = 16; off; off >>= 1) part += __shfl_xor(part, off, 32);

    if (lane == 0) {
        const float s  = part * (1.0f / 512.0f);
        const float sq = s * s;
        out[blockIdx.x] = sq / (1.0f + sq) * (s / (sqrtf(sq + EPS) + EPS));
    }
}

// ---------------------------------------------------------------------------
extern "C" void kernel_launch(void* const* d_in, const int* in_sizes, int n_in,
                              void* d_out, int out_size, void* d_ws, size_t ws_size,
                              hipStream_t stream) {
    const float* x      = (const float*)d_in[0];  // (1,3,24,24)
    const float* conv_w = (const float*)d_in[1];  // (128,3,9,9)
    const float* conv_b = (const float*)d_in[2];  // (128,)
    const float* pri_w  = (const float*)d_in[3];  // (256,128,9,9)
    const float* pri_b  = (const float*)d_in[4];  // (256,)
    const float* W      = (const float*)d_in[5];  // (1,512,10,16,8)
    float* out = (float*)d_out;                   // 160 fp32

    float* ws = (float*)d_ws;
    float* h  = ws + WS_H;
    float* u  = ws + WS_U;

    k_stage1<<<384, 256, 0, stream>>>(x, conv_w, ws);          // 98304 threads
    k1_conv1<<<128, 32, 0, stream>>>(ws, conv_b, h);
    k_im2col2<<<648, 256, 0, stream>>>(h, ws);                 // 165888 threads
    k2_conv2_squash<<<16, 256, 0, stream>>>(ws, pri_w, pri_b, u);
    k3_route<<<160, 32, 0, stream>>>(W, u, out);
}